// ProjectionLayer_55327768707126
// MI455X (gfx1250) — compile-verified
//
#include <hip/hip_runtime.h>
#include <math.h>

// Shapes fixed by the reference setup_inputs().
#define B_ 8
#define S_ 4096
#define H_ 1024
#define M_ 256
#define N_ (B_ * M_)   // 2048 cell rows

typedef __attribute__((ext_vector_type(2))) float v2f;
typedef __attribute__((ext_vector_type(8))) float v8f;

// ---------------------------------------------------------------------------
// Kernel 1: zero the segment-sum accumulator (float4 stores).
// ---------------------------------------------------------------------------
__global__ void zero_ws_kernel(float* __restrict__ p, int n4) {
    int i = blockIdx.x * blockDim.x + threadIdx.x;
    if (i < n4) {
        ((float4*)p)[i] = make_float4(0.f, 0.f, 0.f, 0.f);
    }
}

// ---------------------------------------------------------------------------
// Kernel 2: segment-sum scatter. One block (256 threads) per token.
// Coalesced float4 read of hidden_states, 4x global_atomic_add_f32 into cell.
// ---------------------------------------------------------------------------
__global__ __launch_bounds__(256) void scatter_add_kernel(
    const float* __restrict__ hs, const int* __restrict__ ids,
    float* __restrict__ cell) {
    int tok = blockIdx.x;             // 0 .. B*S-1
    int b   = tok >> 12;              // S = 4096
    int id  = ids[tok];               // 0 .. M-1
    const float4* src = (const float4*)(hs + (long)tok * H_);
    float* dst = cell + (long)(b * M_ + id) * H_;

    int t = threadIdx.x;              // 0..255, H/4 = 256 float4 per row
    float4 v = src[t];
    int h = t * 4;
    atomicAdd(dst + h + 0, v.x);
    atomicAdd(dst + h + 1, v.y);
    atomicAdd(dst + h + 2, v.z);
    atomicAdd(dst + h + 3, v.w);
}

// ---------------------------------------------------------------------------
// Kernel 3: C[n,o] = gelu( sum_k A[n,k] * W[o,k] + bias[o] )
// FP32 WMMA (v_wmma_f32_16x16x4_f32), LDS-tiled.
//   Block tile: 128 (n) x 128 (o), 256 threads = 8 waves (wave32).
//   Wave tile:  64 (n) x 32 (o)  -> 4x2 accumulator tiles of 16x16.
//   K chunk:    32, staged in LDS with row stride 36 floats
//               (36*4 = 144 B: 16B-aligned rows; 9 odd -> 16 distinct banks).
// Fragment layouts per CDNA5 ISA 7.12.2:
//   A 16x4 f32: lane l holds {A[l&15, k+2*(l>>4)], A[l&15, k+2*(l>>4)+1]}
//   B 4x16:     lane l holds {B[k+2*(l>>4), l&15], B[k+2*(l>>4)+1, l&15]}
//               and B[k,o] = W[o,k] (K-contiguous) -> same float2 pattern.
// ---------------------------------------------------------------------------
#define KC  32
#define LDA 36
#define BN  128
#define BO  128

__global__ __launch_bounds__(256) void gemm_gelu_kernel(
    const float* __restrict__ A,     // [N_, H_] cell
    const float* __restrict__ W,     // [H_, H_] weight, row o contiguous in k
    const float* __restrict__ bias,  // [H_]
    float* __restrict__ C) {         // [N_, H_] gelu output
    __shared__ float sA[BN * LDA];   // 18.0 KB
    __shared__ float sB[BO * LDA];   // 18.0 KB

    const int t    = threadIdx.x;
    const int n0   = blockIdx.x * BN;
    const int o0   = blockIdx.y * BO;
    const int wid  = t >> 5;
    const int lane = t & 31;
    const int wn   = (wid & 1) * 64;   // wave row offset inside block tile
    const int wo   = (wid >> 1) * 32;  // wave col offset inside block tile
    const int lrow = lane & 15;
    const int koff = (lane >> 4) * 2;  // 0 or 2

    v8f acc[4][2];
#pragma unroll
    for (int mi = 0; mi < 4; ++mi)
#pragma unroll
        for (int oi = 0; oi < 2; ++oi) {
            v8f z = {};
            acc[mi][oi] = z;
        }

    for (int k0 = 0; k0 < H_; k0 += KC) {
        // Stage A and W chunks: 128 rows x 32 cols each, 1024 float4 per array.
#pragma unroll
        for (int i = 0; i < 4; ++i) {
            int q  = t + i * 256;      // float4 index 0..1023
            int r  = q >> 3;           // 8 float4 per 32-float row
            int c4 = (q & 7) * 4;
            float4 av = *(const float4*)(A + (long)(n0 + r) * H_ + k0 + c4);
            *(float4*)(&sA[r * LDA + c4]) = av;
            float4 bv = *(const float4*)(W + (long)(o0 + r) * H_ + k0 + c4);
            *(float4*)(&sB[r * LDA + c4]) = bv;
        }
        __syncthreads();

#pragma unroll
        for (int k4 = 0; k4 < KC; k4 += 4) {
            v2f af[4];
            v2f bf[2];
#pragma unroll
            for (int mi = 0; mi < 4; ++mi)
                af[mi] = *(const v2f*)(&sA[(wn + mi * 16 + lrow) * LDA + k4 + koff]);
#pragma unroll
            for (int oi = 0; oi < 2; ++oi)
                bf[oi] = *(const v2f*)(&sB[(wo + oi * 16 + lrow) * LDA + k4 + koff]);
#pragma unroll
            for (int mi = 0; mi < 4; ++mi)
#pragma unroll
                for (int oi = 0; oi < 2; ++oi)
                    acc[mi][oi] = __builtin_amdgcn_wmma_f32_16x16x4_f32(
                        /*neg_a=*/false, af[mi],
                        /*neg_b=*/false, bf[oi],
                        /*c_mod=*/(short)0, acc[mi][oi],
                        /*reuse_a=*/false, /*reuse_b=*/false);
        }
        __syncthreads();
    }

    // Epilogue: bias + exact erf-GELU, per C/D layout:
    //   VGPR v: lanes 0-15 -> row v,      col = lane
    //           lanes16-31 -> row v + 8,  col = lane - 16
    const int mhalf = (lane >> 4) * 8;
#pragma unroll
    for (int mi = 0; mi < 4; ++mi) {
#pragma unroll
        for (int oi = 0; oi < 2; ++oi) {
            int o = o0 + wo + oi * 16 + lrow;
            float bo = bias[o];
#pragma unroll
            for (int v = 0; v < 8; ++v) {
                int n = n0 + wn + mi * 16 + mhalf + v;
                float x = acc[mi][oi][v] + bo;
                float g = 0.5f * x * (1.0f + erff(x * 0.70710678118654752f));
                C[(long)n * H_ + o] = g;
            }
        }
    }
}

// ---------------------------------------------------------------------------
// Kernel 4: gather rows back to sequence positions. One block per token.
// ---------------------------------------------------------------------------
__global__ __launch_bounds__(256) void gather_out_kernel(
    const float* __restrict__ cellg, const int* __restrict__ ids,
    float* __restrict__ out) {
    int tok = blockIdx.x;
    int b   = tok >> 12;
    int id  = ids[tok];
    const float4* src = (const float4*)(cellg + (long)(b * M_ + id) * H_);
    float4* dst = (float4*)(out + (long)tok * H_);
    dst[threadIdx.x] = src[threadIdx.x];
}

// ---------------------------------------------------------------------------
// Host launcher. d_in order: hidden_states, W, b, token_type_ids, max_length.
// Workspace: cell (8 MB) + cellg (8 MB) in d_ws.
// ---------------------------------------------------------------------------
extern "C" void kernel_launch(void* const* d_in, const int* in_sizes, int n_in,
                              void* d_out, int out_size, void* d_ws, size_t ws_size,
                              hipStream_t stream) {
    const float* hs   = (const float*)d_in[0];
    const float* W    = (const float*)d_in[1];
    const float* bias = (const float*)d_in[2];
    const int*   ids  = (const int*)d_in[3];
    float* out  = (float*)d_out;

    float* cell  = (float*)d_ws;
    float* cellg = cell + (size_t)N_ * H_;

    const int cell_f4 = N_ * H_ / 4;  // 524288 float4
    zero_ws_kernel<<<(cell_f4 + 255) / 256, 256, 0, stream>>>(cell, cell_f4);

    scatter_add_kernel<<<B_ * S_, 256, 0, stream>>>(hs, ids, cell);

    dim3 grid(N_ / BN, H_ / BO);      // 16 x 8 blocks
    gemm_gelu_kernel<<<grid, 256, 0, stream>>>(cell, W, bias, cellg);

    gather_out_kernel<<<B_ * S_, 256, 0, stream>>>(cellg, ids, out);
}